// HierarchicalLossNetwork_4011499454986
// MI455X (gfx1250) — compile-verified
//
#include <hip/hip_runtime.h>
#include <hip/hip_bf16.h>
#include <math.h>

typedef float v2f __attribute__((ext_vector_type(2)));
typedef float v8f __attribute__((ext_vector_type(8)));

#define BATCH 16384

// ---------------------------------------------------------------------------
// Per-level: one wave (32 lanes) per row. Single-pass online softmax + argmax.
// Lane-strided columns -> fully coalesced 128B reads; logits read exactly once.
// Deterministic shuffle-tree wave reduction.
// ---------------------------------------------------------------------------
__global__ __launch_bounds__(256) void hln_level_kernel(
    const float* __restrict__ pred, const int* __restrict__ labels,
    float* __restrict__ logp_out, int* __restrict__ amax_out,
    int C, int level)
{
    const int wave = threadIdx.x >> 5;
    const int lane = threadIdx.x & 31;
    const int row  = blockIdx.x * 8 + wave;
    if (row >= BATCH) return;
    const float* x = pred + (size_t)row * (size_t)C;

    // online softmax state per lane (C >= 64, so every lane sees >= 2 elems)
    float m = -INFINITY;
    float s = 0.0f;
    int   idx = 0;
    for (int col = lane; col < C; col += 32) {
        float v = x[col];
        if (v > m) {                      // strict '>' keeps first occurrence
            s = s * expf(m - v) + 1.0f;   // expf(-inf)=0 handles first elem
            m = v; idx = col;
        } else {
            s += expf(v - m);
        }
    }
    // deterministic butterfly reduction; lane 0 ends with the full result
    for (int off = 16; off > 0; off >>= 1) {
        float m2 = __shfl_down(m,   off, 32);
        int   i2 = __shfl_down(idx, off, 32);
        float s2 = __shfl_down(s,   off, 32);
        float mn = fmaxf(m, m2);
        float sn = s * expf(m - mn) + s2 * expf(m2 - mn);
        int   ir;
        if      (m2 > m) ir = i2;
        else if (m > m2) ir = idx;
        else             ir = (i2 < idx) ? i2 : idx;   // tie: lowest index
        m = mn; s = sn; idx = ir;
    }
    if (lane == 0) {
        int   lab = labels[row * 4 + level];
        float xl  = x[lab];
        logp_out[row] = xl - m - logf(s);   // log_softmax at label
        amax_out[row] = idx;
    }
}

// ---------------------------------------------------------------------------
// Dependence term per row: D = adj[p][c] (0 if p==0 or c==0);
// term = 3^(D*lp) * 3^(D*lc) - 1  summed over levels 1..3.
// ---------------------------------------------------------------------------
__global__ __launch_bounds__(256) void hln_dep_kernel(
    const int* __restrict__ amax, const int* __restrict__ labels,
    const int* __restrict__ adj1, const int* __restrict__ adj2,
    const int* __restrict__ adj3, float* __restrict__ dterm)
{
    const int row = blockIdx.x * blockDim.x + threadIdx.x;
    if (row >= BATCH) return;

    float t = 0.0f;
    // l = 1: prev C=64, curr C=256
    {
        int p = amax[0 * BATCH + row], c = amax[1 * BATCH + row];
        int D = (p == 0 || c == 0) ? 0 : adj1[p * 256 + c];
        int lp = (p != labels[row * 4 + 0]);
        int lc = (c != labels[row * 4 + 1]);
        t += ((D && lp) ? 3.0f : 1.0f) * ((D && lc) ? 3.0f : 1.0f) - 1.0f;
    }
    // l = 2: prev C=256, curr C=1024
    {
        int p = amax[1 * BATCH + row], c = amax[2 * BATCH + row];
        int D = (p == 0 || c == 0) ? 0 : adj2[p * 1024 + c];
        int lp = (p != labels[row * 4 + 1]);
        int lc = (c != labels[row * 4 + 2]);
        t += ((D && lp) ? 3.0f : 1.0f) * ((D && lc) ? 3.0f : 1.0f) - 1.0f;
    }
    // l = 3: prev C=1024, curr C=4096
    {
        int p = amax[2 * BATCH + row], c = amax[3 * BATCH + row];
        int D = (p == 0 || c == 0) ? 0 : adj3[p * 4096 + c];
        int lp = (p != labels[row * 4 + 2]);
        int lc = (c != labels[row * 4 + 3]);
        t += ((D && lp) ? 3.0f : 1.0f) * ((D && lc) ? 3.0f : 1.0f) - 1.0f;
    }
    dterm[row] = t;
}

// ---------------------------------------------------------------------------
// Final scalar reduction with V_WMMA_F32_16X16X4_F32 (B = ones):
// D[i][j] = sum_k A[i][k]  -> each wmma sums 64 contiguous fp32 values with a
// fixed hardware accumulation order (deterministic, atomic-free).
// A 16x4 f32 layout: lane L<16 holds A[L][0..1]; lane L+16 holds A[L][2..3].
// One block (8 waves); each wave owns a 2048-element chunk per array.
// ---------------------------------------------------------------------------
__global__ __launch_bounds__(256) void hln_final_kernel(
    const float* __restrict__ logp, const float* __restrict__ dterm,
    float* __restrict__ out)
{
    __shared__ float smem[8];
    const int wave = threadIdx.x >> 5;
    const int lane = threadIdx.x & 31;
    const int l16  = lane & 15;
    const int hsel = (lane >> 4) ? 2 : 0;

    const float* arrs[5] = { logp, logp + BATCH, logp + 2 * BATCH,
                             logp + 3 * BATCH, dterm };
    // CE weight: alpha_l * (-1/B); dloss weight: BETA
    const float w[5] = { -0.5f    / 16384.0f,
                         -0.65f   / 16384.0f,
                         -0.845f  / 16384.0f,
                         -1.0985f / 16384.0f,
                          0.8f };

    const int chunk = BATCH / 8;   // 2048 per wave
    const int iters = chunk / 64;  // 32 wmma per array per wave
    v2f bones; bones.x = 1.0f; bones.y = 1.0f;

    float acc = 0.0f;
    for (int a = 0; a < 5; ++a) {
        const float* v = arrs[a] + wave * chunk;
        v8f c = {0.0f, 0.0f, 0.0f, 0.0f, 0.0f, 0.0f, 0.0f, 0.0f};
        for (int it = 0; it < iters; ++it) {
            int off = it * 64 + 4 * l16 + hsel;
            v2f av; av.x = v[off]; av.y = v[off + 1];
            // D = A(16x4) * ones(4x16) + C  -> row sums replicated per column
            c = __builtin_amdgcn_wmma_f32_16x16x4_f32(
                    false, av, false, bones, (short)0, c, false, false);
        }
        float s = c[0] + c[1] + c[2] + c[3] + c[4] + c[5] + c[6] + c[7];
        // lanes 0..15 hold sum over rows 0..7, lanes 16..31 over rows 8..15
        float tot = __shfl(s, 0, 32) + __shfl(s, 16, 32);
        acc += w[a] * tot;
    }
    if (lane == 0) smem[wave] = acc;
    __syncthreads();
    if (threadIdx.x == 0) {
        float t = 0.0f;
        for (int i = 0; i < 8; ++i) t += smem[i];   // fixed order
        out[0] = t;
    }
}

// ---------------------------------------------------------------------------
extern "C" void kernel_launch(void* const* d_in, const int* in_sizes, int n_in,
                              void* d_out, int out_size, void* d_ws, size_t ws_size,
                              hipStream_t stream) {
    const float* pred0  = (const float*)d_in[0];
    const float* pred1  = (const float*)d_in[1];
    const float* pred2  = (const float*)d_in[2];
    const float* pred3  = (const float*)d_in[3];
    const int*   labels = (const int*)d_in[4];
    const int*   adj1   = (const int*)d_in[5];
    const int*   adj2   = (const int*)d_in[6];
    const int*   adj3   = (const int*)d_in[7];
    float*       out    = (float*)d_out;

    // workspace layout: logp[4*B] f32 | amax[4*B] i32 | dterm[B] f32
    float* ws_logp  = (float*)d_ws;
    int*   ws_amax  = (int*)((char*)d_ws + (size_t)4 * BATCH * sizeof(float));
    float* ws_dterm = (float*)((char*)d_ws + (size_t)8 * BATCH * sizeof(float));

    const dim3 blk(256);
    const dim3 grd_rows(BATCH / 8);      // 8 rows (waves) per block

    hln_level_kernel<<<grd_rows, blk, 0, stream>>>(pred0, labels,
        ws_logp + 0 * BATCH, ws_amax + 0 * BATCH, 64, 0);
    hln_level_kernel<<<grd_rows, blk, 0, stream>>>(pred1, labels,
        ws_logp + 1 * BATCH, ws_amax + 1 * BATCH, 256, 1);
    hln_level_kernel<<<grd_rows, blk, 0, stream>>>(pred2, labels,
        ws_logp + 2 * BATCH, ws_amax + 2 * BATCH, 1024, 2);
    hln_level_kernel<<<grd_rows, blk, 0, stream>>>(pred3, labels,
        ws_logp + 3 * BATCH, ws_amax + 3 * BATCH, 4096, 3);

    hln_dep_kernel<<<BATCH / 256, blk, 0, stream>>>(ws_amax, labels,
        adj1, adj2, adj3, ws_dterm);

    hln_final_kernel<<<1, blk, 0, stream>>>(ws_logp, ws_dterm, out);
}